// AttentionHead_70153995812899
// MI455X (gfx1250) — compile-verified
//
#include <hip/hip_runtime.h>
#include <hip/hip_bf16.h>

#define N_NODES 50000
#define N_EDGES 800000
#define IN_DIM  256
#define OUT_DIM 128

typedef float v2f __attribute__((ext_vector_type(2)));
typedef float v8f __attribute__((ext_vector_type(8)));

// ---------------------------------------------------------------------------
// Kernel 1: h' = X @ W with f32 WMMA (V_WMMA_F32_16X16X4_F32), fused with the
// per-node logit GEMVs e_dst = h'@a_dst+b_dst, e_src = h'@a_src+b_src.
// Block = 256 threads = 8 waves; each wave computes a 16x128 row-slab as 8
// 16x16 tiles. W staged in LDS in two 64KB halves, K-pair interleaved so each
// B fragment is a single aligned ds_load_b64.
// ---------------------------------------------------------------------------
__global__ __launch_bounds__(256) void gat_gemm(const float* __restrict__ X,
                                                const float* __restrict__ W,
                                                const float* __restrict__ a_dst,
                                                const float* __restrict__ b_dst,
                                                const float* __restrict__ a_src,
                                                const float* __restrict__ b_src,
                                                float* __restrict__ HP,
                                                float* __restrict__ e_dstv,
                                                float* __restrict__ e_srcv) {
  // interleaved: wlds[(k>>1)*256 + col*2 + (k&1)], 64 pair-rows per phase
  __shared__ alignas(16) float wlds[64 * 256];  // 64 KB

  const int tid   = threadIdx.x;
  const int lane  = tid & 31;
  const int wave  = tid >> 5;
  const int m0    = blockIdx.x * 128 + wave * 16;
  const int rsel  = lane & 15;   // A row / B,C col within tile
  const int khalf = lane >> 4;   // 0: K pair {0,1}; 1: K pair {2,3}

  // clamped A row pointer: OOB rows compute finite garbage, never stored
  int arow = m0 + rsel;
  if (arow >= N_NODES) arow = N_NODES - 1;
  const float* xrow = X + (size_t)arow * IN_DIM;

  // prefetch the phase-2 half of W into L2/WGP$ while phase 1 runs
#pragma unroll
  for (int i = 0; i < 2; ++i)
    __builtin_prefetch(W + 128 * OUT_DIM + (tid * 2 + i) * 32, 0, 1);

  v8f acc[8] = {};

  for (int p = 0; p < 2; ++p) {
    // cooperative interleaved load of W K-rows [p*128, p*128+128) into LDS
    const float* wg = W + (size_t)p * 128 * OUT_DIM;
    for (int i = tid; i < 64 * 128; i += 256) {
      const int r = i >> 7;        // pair-row
      const int c = i & 127;       // column
      v2f v;
      v.x = wg[(2 * r) * OUT_DIM + c];
      v.y = wg[(2 * r + 1) * OUT_DIM + c];
      *(v2f*)&wlds[(size_t)r * 256 + c * 2] = v;
    }
    __syncthreads();

#pragma unroll 4
    for (int kl = 0; kl < 128; kl += 4) {
      const int kg = p * 128 + kl + khalf * 2;
      // A fragment (16x4 f32): lane holds X[arow][kg], X[arow][kg+1]
      v2f a;
      a.x = xrow[kg];
      a.y = xrow[kg + 1];

      const float* wb0 = &wlds[(size_t)(kl / 2 + khalf) * 256 + rsel * 2];
#pragma unroll
      for (int t = 0; t < 8; ++t) {
        // B fragment (4x16 f32): contiguous pair -> one ds_load_b64
        const v2f b = *(const v2f*)&wb0[t * 32];
        acc[t] = __builtin_amdgcn_wmma_f32_16x16x4_f32(
            false, a, false, b, (short)0, acc[t], false, false);
      }
    }
    __syncthreads();
  }

  // --- epilogue 1: store h'. C layout: VGPR j, lane: row = m0+j+khalf*8,
  // col = t*16 + rsel
#pragma unroll
  for (int j = 0; j < 8; ++j) {
    const int row = m0 + j + khalf * 8;
    if (row < N_NODES) {
      float* orow = HP + (size_t)row * OUT_DIM + rsel;
#pragma unroll
      for (int t = 0; t < 8; ++t) orow[t * 16] = acc[t][j];
    }
  }

  // --- epilogue 2: fused logit GEMVs from the live C registers
  float ad[8], as[8];
#pragma unroll
  for (int t = 0; t < 8; ++t) {
    ad[t] = a_dst[t * 16 + rsel];
    as[t] = a_src[t * 16 + rsel];
  }
  const float bd = b_dst[0], bs = b_src[0];
#pragma unroll
  for (int j = 0; j < 8; ++j) {
    float pd = 0.0f, ps = 0.0f;
#pragma unroll
    for (int t = 0; t < 8; ++t) {
      pd += acc[t][j] * ad[t];
      ps += acc[t][j] * as[t];
    }
    // reduce over the 16 lanes of each half (xor 1/2/4/8 stays within half)
#pragma unroll
    for (int off = 8; off >= 1; off >>= 1) {
      pd += __shfl_xor(pd, off, 32);
      ps += __shfl_xor(ps, off, 32);
    }
    const int row = m0 + j + khalf * 8;
    if (rsel == 0 && row < N_NODES) {
      e_dstv[row] = pd + bd;
      e_srcv[row] = ps + bs;
    }
  }
}

// ---------------------------------------------------------------------------
// Kernel 2: init accumulators (every launch -> deterministic/replay-safe)
// ---------------------------------------------------------------------------
__global__ void gat_init(float* __restrict__ out_acc,
                         unsigned* __restrict__ max_enc,
                         float* __restrict__ sum) {
  const int i = blockIdx.x * blockDim.x + threadIdx.x;
  if (i < N_NODES * OUT_DIM) out_acc[i] = 0.0f;
  if (i < N_NODES) {
    max_enc[i] = 0x00800000u;  // order-encoding of -FLT_MAX
    sum[i] = 0.0f;
  }
}

// ---------------------------------------------------------------------------
// Kernel 3: per-edge leaky-relu scores + segment max via uint-ordered atomicMax
// ---------------------------------------------------------------------------
__global__ void gat_scores(const int* __restrict__ esrc, const int* __restrict__ edst,
                           const float* __restrict__ e_dst, const float* __restrict__ e_src,
                           float* __restrict__ scores, unsigned* __restrict__ max_enc) {
  const int e = blockIdx.x * blockDim.x + threadIdx.x;
  if (e >= N_EDGES) return;
  const int d = edst[e], s = esrc[e];
  float v = e_dst[d] + e_src[s];
  v = (v >= 0.0f) ? v : 0.2f * v;
  scores[e] = v;
  unsigned u = __float_as_uint(v);
  u = (u & 0x80000000u) ? ~u : (u | 0x80000000u);  // order-preserving map
  atomicMax(&max_enc[d], u);
}

// ---------------------------------------------------------------------------
// Kernel 4: exp(score - max[dst]) + segment sum
// ---------------------------------------------------------------------------
__global__ void gat_exp(const int* __restrict__ edst,
                        float* __restrict__ scores,
                        const unsigned* __restrict__ max_enc,
                        float* __restrict__ sum) {
  const int e = blockIdx.x * blockDim.x + threadIdx.x;
  if (e >= N_EDGES) return;
  const int d = edst[e];
  const unsigned u = max_enc[d];
  const float m = (u & 0x80000000u) ? __uint_as_float(u ^ 0x80000000u)
                                    : __uint_as_float(~u);
  const float ex = expf(scores[e] - m);
  scores[e] = ex;  // reuse buffer for exp_scores
  atomicAdd(&sum[d], ex);
}

// ---------------------------------------------------------------------------
// Kernel 5: weighted scatter-add of messages. One wave per edge; float4 gather
// of h'[src]; f32 atomic adds into out[dst] (resolve in the 192 MB L2).
// ---------------------------------------------------------------------------
__global__ __launch_bounds__(256) void gat_aggregate(const int* __restrict__ esrc,
                                                     const int* __restrict__ edst,
                                                     const float* __restrict__ expsc,
                                                     const float* __restrict__ sum,
                                                     const float* __restrict__ HP,
                                                     float* __restrict__ out_acc) {
  const int lane = threadIdx.x & 31;
  const int e = blockIdx.x * 8 + (threadIdx.x >> 5);
  if (e >= N_EDGES) return;
  const int d = edst[e], s = esrc[e];
  const float alpha = expsc[e] / fmaxf(sum[d], 1e-12f);
  const float4 h = ((const float4*)(HP + (size_t)s * OUT_DIM))[lane];
  float* o = out_acc + (size_t)d * OUT_DIM + lane * 4;
  atomicAdd(o + 0, h.x * alpha);
  atomicAdd(o + 1, h.y * alpha);
  atomicAdd(o + 2, h.z * alpha);
  atomicAdd(o + 3, h.w * alpha);
}

// ---------------------------------------------------------------------------
// Kernel 6: out = ELU(acc + bias), in place
// ---------------------------------------------------------------------------
__global__ void gat_finalize(float* __restrict__ out, const float* __restrict__ bias) {
  const int i = blockIdx.x * blockDim.x + threadIdx.x;
  if (i >= N_NODES * OUT_DIM) return;
  const float c = out[i] + bias[i & (OUT_DIM - 1)];
  out[i] = (c > 0.0f) ? c : expm1f(c);
}

// ---------------------------------------------------------------------------
extern "C" void kernel_launch(void* const* d_in, const int* in_sizes, int n_in,
                              void* d_out, int out_size, void* d_ws, size_t ws_size,
                              hipStream_t stream) {
  const float* X      = (const float*)d_in[0];
  const int*   esrc   = (const int*)d_in[1];
  const int*   edst   = (const int*)d_in[2];
  const float* W      = (const float*)d_in[3];
  const float* a_dst  = (const float*)d_in[4];
  const float* b_dst  = (const float*)d_in[5];
  const float* a_src  = (const float*)d_in[6];
  const float* b_src  = (const float*)d_in[7];
  const float* obias  = (const float*)d_in[8];
  float* out = (float*)d_out;

  char* ws = (char*)d_ws;
  float*    HP      = (float*)(ws);               // 25.6 MB
  float*    e_dstv  = (float*)(ws + 25600000);    // 200 KB
  float*    e_srcv  = (float*)(ws + 25800000);    // 200 KB
  unsigned* max_enc = (unsigned*)(ws + 26000000); // 200 KB
  float*    sumv    = (float*)(ws + 26200000);    // 200 KB
  float*    scores  = (float*)(ws + 26400000);    // 3.2 MB

  // 1) GEMM h' = X @ W (f32 WMMA) + fused per-node logits
  gat_gemm<<<(N_NODES + 127) / 128, 256, 0, stream>>>(X, W, a_dst, b_dst,
                                                      a_src, b_src,
                                                      HP, e_dstv, e_srcv);
  // 2) init accumulators (every call)
  gat_init<<<(N_NODES * OUT_DIM + 255) / 256, 256, 0, stream>>>(out, max_enc, sumv);
  // 3) per-edge scores + segment max
  gat_scores<<<(N_EDGES + 255) / 256, 256, 0, stream>>>(esrc, edst, e_dstv, e_srcv,
                                                        scores, max_enc);
  // 4) exp + segment sum
  gat_exp<<<(N_EDGES + 255) / 256, 256, 0, stream>>>(edst, scores, max_enc, sumv);
  // 5) weighted scatter-add
  gat_aggregate<<<N_EDGES / 8, 256, 0, stream>>>(esrc, edst, scores, sumv, HP, out);
  // 6) bias + ELU
  gat_finalize<<<(N_NODES * OUT_DIM + 255) / 256, 256, 0, stream>>>(out, obias);
}